// CustomLoss_69999376990919
// MI455X (gfx1250) — compile-verified
//
#include <hip/hip_runtime.h>
#include <hip/hip_bf16.h>

// ---- problem constants (from reference) ----
#define BATCH 512
#define CCH   4
#define NN    16384
#define NCH   (NN / 512)         // 32 chunks of 512 n-values
#define H2    6.103515625e-05f   // (1/128)^2 = 2^-14
#define PSTR  520                // padded LDS row stride for preds (bank spread)

typedef __attribute__((ext_vector_type(2))) float v2f;
typedef __attribute__((ext_vector_type(8))) float v8f;

// CDNA5 async global->LDS copies (ASYNCcnt). VDST VGPR = LDS byte offset;
// low 32 bits of a flat pointer into __shared__ are exactly that offset.
__device__ __forceinline__ void async_cp16(const float* g, float* l) {
    unsigned lds_off = (unsigned)(unsigned long long)l;
    unsigned long long gaddr = (unsigned long long)g;
    asm volatile("global_load_async_to_lds_b128 %0, %1, off"
                 :: "v"(lds_off), "v"(gaddr) : "memory");
}
__device__ __forceinline__ void async_cp8(const float* g, float* l) {
    unsigned lds_off = (unsigned)(unsigned long long)l;
    unsigned long long gaddr = (unsigned long long)g;
    asm volatile("global_load_async_to_lds_b64 %0, %1, off"
                 :: "v"(lds_off), "v"(gaddr) : "memory");
}
// Async loads complete in order per wave (ISA 08 §4.1): waiting to <=5 while 5
// new copies are in flight guarantees the previous chunk's 5 copies landed.
__device__ __forceinline__ void wait_async_le5() {
    asm volatile("s_wait_asynccnt 0x5" ::: "memory");
}
__device__ __forceinline__ void wait_async_0() {
    asm volatile("s_wait_asynccnt 0x0" ::: "memory");
}

// One block per batch sample. 256 threads = 8 wave32.
// [G|W](4x8) = (a*V)(4xN) . [V^T | T](Nx8)  via v_wmma_f32_16x16x4_f32,
// K = N in chunks of 4. Double-buffered LDS staging driven by async DMA so
// chunk k+1's copies overlap chunk k's WMMA work. H^2 applied in the epilogue.
__global__ __launch_bounds__(256)
void gramloss_kernel(const float* __restrict__ Cf,   // coefficients [B][N]
                     const float* __restrict__ P,    // predictions  [B][C][N]
                     const float* __restrict__ T,    // targets      [B][N][C]
                     float* __restrict__ ws)         // per-batch loss [B]
{
    const int b    = blockIdx.x;
    const int tid  = threadIdx.x;
    const int lane = tid & 31;
    const int wid  = tid >> 5;

    __shared__ __align__(16) float lds_p[2][4 * PSTR];  // preds rows, padded stride
    __shared__ __align__(16) float lds_t[2][512 * 4];   // targets [n][m]
    __shared__ __align__(16) float lds_a[2][512];       // raw coefficients
    __shared__ float lds_red[8 * 32];                   // per-wave 4x8 partials
    __shared__ float lds_fin[32];                       // reduced [G|W] (unscaled)

    const float* Pb = P + (size_t)b * CCH * NN;
    const float* Tb = T + (size_t)b * NN * CCH;
    const float* Cb = Cf + (size_t)b * NN;

    v8f acc = {};                 // 16x16 f32 accumulator (rows 0-3 / cols 0-7 useful)
    const int row = lane & 15;    // A-matrix M index / B-matrix N index (layout §7.12.2)
    const int kh  = lane >> 4;    // K half: lanes 0-15 -> K=0,1 ; lanes 16-31 -> K=2,3

    // Per-thread staging assignments (constant across chunks):
    const int j0 = tid,        c0 = j0 >> 7, i0s = j0 & 127;   // preds float4 #0
    const int j1 = tid + 256,  c1 = j1 >> 7, i1s = j1 & 127;   // preds float4 #1

    // Stage one 512-n chunk into buffer `buf`: exactly 5 async ops per wave.
    auto stage = [&](int chunk, int buf) {
        const int base = chunk << 9;
        async_cp16(Pb + (size_t)c0 * NN + base + i0s * 4, &lds_p[buf][c0 * PSTR + i0s * 4]);
        async_cp16(Pb + (size_t)c1 * NN + base + i1s * 4, &lds_p[buf][c1 * PSTR + i1s * 4]);
        async_cp16(Tb + (size_t)base * 4 + j0 * 4,        &lds_t[buf][j0 * 4]);
        async_cp16(Tb + (size_t)base * 4 + j1 * 4,        &lds_t[buf][j1 * 4]);
        async_cp8 (Cb + base + tid * 2,                   &lds_a[buf][tid * 2]);
    };

    stage(0, 0);                  // prologue: fill buffer 0

    for (int chunk = 0; chunk < NCH; ++chunk) {
        const int cur = chunk & 1;

        // Issue next chunk's DMA into the other buffer (its readers finished at
        // the previous iteration's end-barrier), then wait for current chunk.
        if (chunk + 1 < NCH) {
            stage(chunk + 1, cur ^ 1);

            // L2 prefetch two chunks ahead (global_prefetch_b8).
            const int base2 = (chunk + 2) << 9;
            if (base2 < NN) {
                if (tid < 64)
                    __builtin_prefetch(Pb + (size_t)(tid >> 4) * NN + base2 + (tid & 15) * 32, 0, 3);
                else if (tid < 128)
                    __builtin_prefetch(Tb + (size_t)base2 * 4 + (tid - 64) * 32, 0, 3);
                else if (tid < 144)
                    __builtin_prefetch(Cb + base2 + (tid - 128) * 32, 0, 3);
            }
            wait_async_le5();     // current chunk landed; next chunk stays in flight
        } else {
            wait_async_0();
        }
        __syncthreads();          // all waves' DMAs for `cur` visible

        // ---- each wave: 64 n-values -> 16 WMMA K-steps of 4 ----
        const float* lp = lds_p[cur];
        const float* lt = lds_t[cur];
        const float* la = lds_a[cur];
        const int s0 = wid << 6;
        #pragma unroll
        for (int t = 0; t < 16; ++t) {
            const int i0 = s0 + (t << 2) + (kh << 1);
            const int i1 = i0 + 1;
            const float a0 = la[i0];
            const float a1 = la[i1];

            v2f A, Bm;
            // A (16x4): rows 0-3 = coeff-weighted prediction rows, rest zero
            A.x = (row < 4) ? a0 * lp[row * PSTR + i0] : 0.0f;
            A.y = (row < 4) ? a1 * lp[row * PSTR + i1] : 0.0f;
            // B (4x16): cols 0-3 = raw prediction rows (-> G), cols 4-7 = target cols (-> W)
            float b0, b1;
            if (row < 4)      { b0 = lp[row * PSTR + i0];    b1 = lp[row * PSTR + i1]; }
            else if (row < 8) { b0 = lt[i0 * 4 + (row - 4)]; b1 = lt[i1 * 4 + (row - 4)]; }
            else              { b0 = 0.0f;                    b1 = 0.0f; }
            Bm.x = b0; Bm.y = b1;

            acc = __builtin_amdgcn_wmma_f32_16x16x4_f32(
                false, A, false, Bm, (short)0, acc, false, false);
        }
        __syncthreads();          // `cur` free for reuse at chunk+2's stage()
    }

    // ---- cross-wave reduction of the 4x8 useful block ----
    // C/D layout: VGPR r holds row M=r for lanes 0-15 (N=lane). Useful:
    // rows 0-3 (acc[0..3]) x cols 0-7 (lanes 0-7).
    if (lane < 8) {
        #pragma unroll
        for (int r = 0; r < 4; ++r)
            lds_red[wid * 32 + r * 8 + lane] = acc[r];
    }
    __syncthreads();
    if (tid < 32) {
        float s = 0.0f;
        #pragma unroll
        for (int w = 0; w < 8; ++w) s += lds_red[w * 32 + tid];
        lds_fin[tid] = s;
    }
    __syncthreads();

    // ---- Gram-space Gram-Schmidt + M = cm*W, one lane (apply H^2 here) ----
    if (tid == 0) {
        float g[4][4], wm[4][4];
        for (int r = 0; r < 4; ++r)
            for (int c = 0; c < 4; ++c) {
                g[r][c]  = H2 * lds_fin[r * 8 + c];
                wm[r][c] = H2 * lds_fin[r * 8 + 4 + c];
            }
        // e_j = sum_k cm[j][k] * v_k ; mirrors reference GS incl. n2<=0 guard
        float cm[4][4] = {};
        for (int j = 0; j < 4; ++j) {
            float cu[4] = {0.f, 0.f, 0.f, 0.f};
            cu[j] = 1.0f;
            for (int i = 0; i < j; ++i) {
                float p = 0.0f;
                for (int k = 0; k <= i; ++k) p += cm[i][k] * g[j][k];   // <v_j, e_i>_a
                for (int k = 0; k <= i; ++k) cu[k] -= p * cm[i][k];
            }
            float n2 = 0.0f;
            for (int k = 0; k <= j; ++k)
                for (int l = 0; l <= j; ++l)
                    n2 += cu[k] * cu[l] * g[k][l];                      // <u,u>_a
            const float sc = (n2 > 0.0f) ? rsqrtf(n2) : 1.0f;
            for (int k = 0; k < 4; ++k) cm[j][k] = sc * cu[k];
        }
        float ss = 0.0f;
        for (int j = 0; j < 4; ++j)
            for (int m = 0; m < 4; ++m) {
                float mv = 0.0f;
                for (int k = 0; k <= j; ++k) mv += cm[j][k] * wm[k][m]; // M[j][m]
                ss += mv * mv;
            }
        ws[b] = (4.0f - ss) * 0.25f;
    }
}

// Deterministic mean over the 512 per-batch losses.
__global__ __launch_bounds__(256)
void finalize_kernel(const float* __restrict__ ws, float* __restrict__ out)
{
    __shared__ float red[256];
    const int t = threadIdx.x;
    red[t] = ws[t] + ws[t + 256];
    __syncthreads();
    for (int s = 128; s > 0; s >>= 1) {
        if (t < s) red[t] += red[t + s];
        __syncthreads();
    }
    if (t == 0) out[0] = red[0] * (1.0f / (float)BATCH);
}

extern "C" void kernel_launch(void* const* d_in, const int* in_sizes, int n_in,
                              void* d_out, int out_size, void* d_ws, size_t ws_size,
                              hipStream_t stream) {
    const float* coeff = (const float*)d_in[0];   // [B][N]
    const float* preds = (const float*)d_in[1];   // [B][C][N]
    const float* targs = (const float*)d_in[2];   // [B][N][C]
    float* out = (float*)d_out;
    float* ws  = (float*)d_ws;                    // BATCH floats of scratch

    gramloss_kernel<<<BATCH, 256, 0, stream>>>(coeff, preds, targs, ws);
    finalize_kernel<<<1, 256, 0, stream>>>(ws, out);
}